// WhatWhereUnPooling2D_51522427683437
// MI455X (gfx1250) — compile-verified
//
#include <hip/hip_runtime.h>

// Max-unpool 2x2 (stride 2, non-overlapping), NCHW.
//   in : x[32,64,64,64] f32, where[32,64,64,64] i32 (argmax slot 0..3 = sy*2+sx)
//   out: [32,64,128,128] f32, out[n,c,2h+sy,2w+sx] = x[n,c,h,w] iff where==sy*2+sx else 0
//
// Pure streaming scatter: ~201 MB of traffic, 0 FLOPs -> HBM-bound (~8.6 us at
// 23.3 TB/s). Strategy: B128 loads/stores, non-temporal cache hints, branchless
// cndmask selects, power-of-two index math.

typedef __attribute__((ext_vector_type(4))) float v4f;
typedef __attribute__((ext_vector_type(4))) int   v4i;

namespace {
constexpr int kN = 32, kC = 64, kH = 64, kW = 64;
constexpr int kW4     = kW / 4;                 // 16 quad-pixel groups per row
constexpr int kGroups = kN * kC * kH * kW4;     // 2,097,152 threads total
constexpr int kOutW   = 2 * kW;                 // 128
constexpr int kOutHW  = 2 * kH * kOutW;         // 16384 floats per (n,c) plane
}

__global__ __launch_bounds__(256) void
WhatWhereUnPooling2D_51522427683437_kernel(const float* __restrict__ x,
                                           const int*   __restrict__ where,
                                           float*       __restrict__ out) {
  const int t = blockIdx.x * 256 + threadIdx.x;   // quad-pixel group index

  // Decompose (all power-of-two): t -> (nc, h, w4)
  const int w4   = t & (kW4 - 1);
  const int rest = t >> 4;
  const int h    = rest & (kH - 1);
  const int nc   = rest >> 6;

  // One B128 load of 4 values + one B128 load of 4 argmax slots (TH=NT).
  const v4f xv = __builtin_nontemporal_load((const v4f*)x + t);
  const v4i wv = __builtin_nontemporal_load((const v4i*)where + t);

  // Branchless one-hot expansion: slot = sy*2 + sx.
  // Pixel j contributes to top-row cols {2j, 2j+1} (sy=0) or bottom row (sy=1).
  v4f t0, t1, b0, b1;  // top cols [0..3],[4..7]; bottom cols [0..3],[4..7]
  t0.x = (wv.x == 0) ? xv.x : 0.0f;
  t0.y = (wv.x == 1) ? xv.x : 0.0f;
  b0.x = (wv.x == 2) ? xv.x : 0.0f;
  b0.y = (wv.x == 3) ? xv.x : 0.0f;

  t0.z = (wv.y == 0) ? xv.y : 0.0f;
  t0.w = (wv.y == 1) ? xv.y : 0.0f;
  b0.z = (wv.y == 2) ? xv.y : 0.0f;
  b0.w = (wv.y == 3) ? xv.y : 0.0f;

  t1.x = (wv.z == 0) ? xv.z : 0.0f;
  t1.y = (wv.z == 1) ? xv.z : 0.0f;
  b1.x = (wv.z == 2) ? xv.z : 0.0f;
  b1.y = (wv.z == 3) ? xv.z : 0.0f;

  t1.z = (wv.w == 0) ? xv.w : 0.0f;
  t1.w = (wv.w == 1) ? xv.w : 0.0f;
  b1.z = (wv.w == 2) ? xv.w : 0.0f;
  b1.w = (wv.w == 3) ? xv.w : 0.0f;

  // Output base: (nc, row 2h, col 8*w4). All stores 16B-aligned.
  float* o = out + (size_t)nc * kOutHW + (size_t)(2 * h) * kOutW + (size_t)(8 * w4);

  __builtin_nontemporal_store(t0, (v4f*)o);
  __builtin_nontemporal_store(t1, (v4f*)(o + 4));
  __builtin_nontemporal_store(b0, (v4f*)(o + kOutW));
  __builtin_nontemporal_store(b1, (v4f*)(o + kOutW + 4));
}

extern "C" void kernel_launch(void* const* d_in, const int* in_sizes, int n_in,
                              void* d_out, int out_size, void* d_ws, size_t ws_size,
                              hipStream_t stream) {
  (void)in_sizes; (void)n_in; (void)out_size; (void)d_ws; (void)ws_size;
  const float* x  = (const float*)d_in[0];
  const int*   wh = (const int*)d_in[1];
  float*       o  = (float*)d_out;

  constexpr int threads = 256;
  constexpr int blocks  = kGroups / threads;  // 8192, exact cover
  hipLaunchKernelGGL(WhatWhereUnPooling2D_51522427683437_kernel,
                     dim3(blocks), dim3(threads), 0, stream, x, wh, o);
}